// CliffordMACE_86603720556983
// MI455X (gfx1250) — compile-verified
//
#include <hip/hip_runtime.h>

typedef __attribute__((ext_vector_type(2))) float v2f;
typedef __attribute__((ext_vector_type(8))) float v8f;

#define NUM_RBF 32
#define CUTOFF 5.0f
#define CDIM 16
#define PI_F 3.14159265358979323846f

// ---------------------------------------------------------------------------
// f32 WMMA: D(16x16) += A(16x4) * B(4x16), full-precision accumulate.
// A layout (ISA 7.12.2, 32-bit A 16x4): lane l<16 -> {K0,K1}, lane l+16 -> {K2,K3}
// B assumed mirrored: K = 2*half + vgpr, N = lane%16.
// C/D: vgpr v -> row v (lanes 0-15) / row v+8 (lanes 16-31), col = lane%16.
// ---------------------------------------------------------------------------
__device__ __forceinline__ v8f wmma16x16x4(v2f a, v2f b, v8f c) {
  return __builtin_amdgcn_wmma_f32_16x16x4_f32(
      /*neg_a=*/false, a, /*neg_b=*/false, b,
      /*c_mod=*/(short)0, c, /*reuse_a=*/false, /*reuse_b=*/false);
}

// ---------------------------------------------------------------------------
// Zero the aggregation buffer (workspace is poisoned, must re-zero every call)
// ---------------------------------------------------------------------------
__global__ void cm_zero_kernel(float4* __restrict__ p, long n4) {
  long i = (long)blockIdx.x * blockDim.x + threadIdx.x;
  long stride = (long)gridDim.x * blockDim.x;
  float4 z = make_float4(0.f, 0.f, 0.f, 0.f);
  for (; i < n4; i += stride) p[i] = z;
}

// ---------------------------------------------------------------------------
// Edge stage: 16 edges per 256-thread block (16 channels per edge).
// Phase 1: 16 threads/edge cooperatively compute the 32 RBF gaussians (2 exps
//          per thread) + unit vector + cosine envelope into LDS.
// Phase 2: thread = (edge, channel): w_e = env * <rbf, W_rbf[:,c]>, grade-1
//          geometric product of h[src] with the unit edge multivector,
//          scatter via global f32 atomics into agg[dst].
// ---------------------------------------------------------------------------
__global__ __launch_bounds__(256) void cm_edge_kernel(
    const float* __restrict__ h, const float* __restrict__ pos,
    const int* __restrict__ ei, const float* __restrict__ W_rbf,
    float* __restrict__ agg, int E)
{
  __shared__ float s_rbf[16][NUM_RBF];
  __shared__ float s_u[16][3];
  __shared__ float s_env[16];
  __shared__ float s_W[NUM_RBF][CDIM];

  const int tid = threadIdx.x;
  for (int i = tid; i < NUM_RBF * CDIM; i += 256)
    s_W[i / CDIM][i % CDIM] = W_rbf[i];

  const float gamma = (NUM_RBF / CUTOFF) * (NUM_RBF / CUTOFF);
  const float dmu = CUTOFF / (float)(NUM_RBF - 1);   // linspace(0,5,32) step

  { // phase 1
    int el = tid & 15;
    int j  = tid >> 4;
    int e = blockIdx.x * 16 + el;
    if (e < E) {
      int src = ei[e], dst = ei[E + e];
      float rx = pos[dst*3+0] - pos[src*3+0];
      float ry = pos[dst*3+1] - pos[src*3+1];
      float rz = pos[dst*3+2] - pos[src*3+2];
      float d = sqrtf(rx*rx + ry*ry + rz*rz + 1e-12f);
      float t0 = d - (float)j * dmu;
      float t1 = d - (float)(j + 16) * dmu;
      s_rbf[el][j]      = __expf(-gamma * t0 * t0);
      s_rbf[el][j + 16] = __expf(-gamma * t1 * t1);
      if (j == 0) {
        float inv = 1.0f / d;
        s_u[el][0] = rx * inv; s_u[el][1] = ry * inv; s_u[el][2] = rz * inv;
        float env = 0.5f * (__cosf(PI_F * d / CUTOFF) + 1.0f);
        s_env[el] = (d < CUTOFF) ? env : 0.0f;
      }
    }
  }
  __syncthreads();

  // phase 2
  int el = tid >> 4;
  int c  = tid & 15;
  int e = blockIdx.x * 16 + el;
  if (e >= E) return;
  int src = ei[e], dst = ei[E + e];

  float w = 0.f;
#pragma unroll
  for (int r = 0; r < NUM_RBF; ++r) w = fmaf(s_rbf[el][r], s_W[r][c], w);
  w *= s_env[el];

  const float4* hp = (const float4*)(h + (long)src * 128 + c * 8);
  float4 h0 = hp[0], h1 = hp[1];
  float a0=h0.x, a1=h0.y, a2=h0.z, a3=h0.w, a4=h1.x, a5=h1.y, a6=h1.z, a7=h1.w;
  float u1 = s_u[el][0], u2 = s_u[el][1], u3 = s_u[el][2];

  // Cl(3,0) product a * (u1 e1 + u2 e2 + u3 e3), shortlex blade order
  float m0 = a1*u1 + a2*u2 + a3*u3;
  float m1 = a0*u1 + a4*u2 + a5*u3;
  float m2 = a0*u2 - a4*u1 + a6*u3;
  float m3 = a0*u3 - a5*u1 - a6*u2;
  float m4 = a1*u2 - a2*u1 + a7*u3;
  float m5 = a1*u3 - a3*u1 - a7*u2;
  float m6 = a2*u3 - a3*u2 + a7*u1;
  float m7 = a4*u3 - a5*u2 + a6*u1;

  float* p = agg + (long)dst * 128 + c * 8;
  atomicAdd(p+0, m0*w); atomicAdd(p+1, m1*w);
  atomicAdd(p+2, m2*w); atomicAdd(p+3, m3*w);
  atomicAdd(p+4, m4*w); atomicAdd(p+5, m5*w);
  atomicAdd(p+6, m6*w); atomicAdd(p+7, m7*w);
}

// ---------------------------------------------------------------------------
// Node stage: one wave = 2 nodes. Rows of the WMMA tile are (node,blade)
// pairs (2 nodes x 8 blades = 16 rows, K = 16 channels done as 4x K=4 WMMA).
// After the WMMA, lane (half,l) holds the full 8-blade multivector of
// out[n0+half][channel=l] (and q likewise) -> per-lane full geometric
// product, contiguous 32B store.
// ---------------------------------------------------------------------------
__global__ __launch_bounds__(256) void cm_node_kernel(
    const float* __restrict__ agg, const float* __restrict__ h,
    const float* __restrict__ W_out, const float* __restrict__ W_sgp,
    float* __restrict__ out, int N)
{
  int lane = threadIdx.x & 31;
  int wave = threadIdx.x >> 5;
  int n0 = (blockIdx.x * 8 + wave) * 2;   // wave-uniform
  if (n0 >= N) return;                    // wave-uniform exit: EXEC stays all-1 for WMMA

  int half = lane >> 4;
  int l    = lane & 15;

  // A-row r = l  ->  node n0 + l/8, blade l%8; elements at channel stride 8
  int nodeA = n0 + (l >> 3);
  if (nodeA >= N) nodeA = N - 1;          // clamp loads, stores predicated below
  const float* pA = agg + (long)nodeA * 128 + (l & 7);
  const float* pH = h   + (long)nodeA * 128 + (l & 7);

  v8f acc_o = {0.f,0.f,0.f,0.f,0.f,0.f,0.f,0.f};
  v8f acc_q = {0.f,0.f,0.f,0.f,0.f,0.f,0.f,0.f};

#pragma unroll
  for (int s = 0; s < 4; ++s) {
    int k0 = 4 * s + 2 * half;
    v2f a, b;
    a.x = pA[(k0 + 0) * 8];
    a.y = pA[(k0 + 1) * 8];
    b.x = W_out[(k0 + 0) * 16 + l];
    b.y = W_out[(k0 + 1) * 16 + l];
    acc_o = wmma16x16x4(a, b, acc_o);
    v2f a2, b2;
    a2.x = pH[(k0 + 0) * 8];
    a2.y = pH[(k0 + 1) * 8];
    b2.x = W_sgp[(k0 + 0) * 16 + l];
    b2.y = W_sgp[(k0 + 1) * 16 + l];
    acc_q = wmma16x16x4(a2, b2, acc_q);
  }

  float o0=acc_o[0], o1=acc_o[1], o2=acc_o[2], o3=acc_o[3],
        o4=acc_o[4], o5=acc_o[5], o6=acc_o[6], o7=acc_o[7];
  float q0=acc_q[0], q1=acc_q[1], q2=acc_q[2], q3=acc_q[3],
        q4=acc_q[4], q5=acc_q[5], q6=acc_q[6], q7=acc_q[7];

  // res = out + out * q   (full Cl(3,0) geometric product, shortlex order)
  float r0 = o0 + (o0*q0 + o1*q1 + o2*q2 + o3*q3 - o4*q4 - o5*q5 - o6*q6 - o7*q7);
  float r1 = o1 + (o0*q1 + o1*q0 - o2*q4 - o3*q5 + o4*q2 + o5*q3 - o6*q7 - o7*q6);
  float r2 = o2 + (o0*q2 + o1*q4 + o2*q0 - o3*q6 - o4*q1 + o5*q7 + o6*q3 + o7*q5);
  float r3 = o3 + (o0*q3 + o1*q5 + o2*q6 + o3*q0 - o4*q7 - o5*q1 - o6*q2 - o7*q4);
  float r4 = o4 + (o0*q4 + o1*q2 - o2*q1 + o3*q7 + o4*q0 - o5*q6 + o6*q5 + o7*q3);
  float r5 = o5 + (o0*q5 + o1*q3 - o2*q7 - o3*q1 + o4*q6 + o5*q0 - o6*q4 - o7*q2);
  float r6 = o6 + (o0*q6 + o1*q7 + o2*q3 - o3*q2 - o4*q5 + o5*q4 + o6*q0 + o7*q1);
  float r7 = o7 + (o0*q7 + o1*q6 - o2*q5 + o3*q4 + o4*q3 - o5*q2 + o6*q1 + o7*q0);

  int nd = n0 + half;
  if (nd < N) {
    float4* po = (float4*)(out + (long)nd * 128 + l * 8);
    po[0] = make_float4(r0, r1, r2, r3);
    po[1] = make_float4(r4, r5, r6, r7);
  }
}

// ---------------------------------------------------------------------------
extern "C" void kernel_launch(void* const* d_in, const int* in_sizes, int n_in,
                              void* d_out, int out_size, void* d_ws, size_t ws_size,
                              hipStream_t stream) {
  const float* h     = (const float*)d_in[0];
  const float* pos   = (const float*)d_in[1];
  const int*   ei    = (const int*)  d_in[2];
  const float* W_rbf = (const float*)d_in[3];
  const float* W_out = (const float*)d_in[4];
  const float* W_sgp = (const float*)d_in[5];

  int N = in_sizes[0] / (CDIM * 8);
  int E = in_sizes[2] / 2;

  float* agg = (float*)d_ws;          // [N][16][8] f32, 25.6 MB
  float* out = (float*)d_out;

  long n4 = (long)N * 128 / 4;
  cm_zero_kernel<<<2048, 256, 0, stream>>>((float4*)agg, n4);
  cm_edge_kernel<<<(E + 15) / 16, 256, 0, stream>>>(h, pos, ei, W_rbf, agg, E);
  cm_node_kernel<<<(N + 15) / 16, 256, 0, stream>>>(agg, h, W_out, W_sgp, out, N);
}